// LinTrans_56324201119969
// MI455X (gfx1250) — compile-verified
//
#include <hip/hip_runtime.h>
#include <hip/hip_bf16.h>
#include <stdint.h>

// ---------------------------------------------------------------------------
// Jc = (Ic - Ac) / (0.8*H + 0.2) + Ac,  Ac = mean of k = n/1000 smallest of H
// 3-level histogram select (10/10/12 bits of monotonic key), then fused
// elementwise pass streamed through LDS via gfx1250 async-to-LDS loads.
// ---------------------------------------------------------------------------

#define ALPHA_C 0.8f
#define EPS_C   0.2f

#if defined(__gfx1250__) && __has_builtin(__builtin_amdgcn_global_load_async_to_lds_b128) && __has_builtin(__builtin_amdgcn_s_wait_asynccnt)
#define USE_ASYNC_LDS 1
#else
#define USE_ASYNC_LDS 0
#endif

__device__ __forceinline__ uint32_t monoKey(float x) {
    uint32_t u = __float_as_uint(x);
    return (u & 0x80000000u) ? ~u : (u | 0x80000000u);
}
__device__ __forceinline__ float keyToFloat(uint32_t key) {
    uint32_t u = (key & 0x80000000u) ? (key ^ 0x80000000u) : ~key;
    return __uint_as_float(u);
}

// ---------------------------------------------------------------------------
__global__ void lt_zero(unsigned* __restrict__ p, int nWords) {
    int i = blockIdx.x * blockDim.x + threadIdx.x;
    if (i < nWords) p[i] = 0u;
}

// ---- level-1: LDS-privatized 1024-bin histogram of top 10 key bits --------
__global__ __launch_bounds__(256) void lt_hist1(const float* __restrict__ H,
                                                long long nv4, long long n,
                                                unsigned* __restrict__ hist1) {
    __shared__ unsigned sh[1024];
    for (int i = threadIdx.x; i < 1024; i += blockDim.x) sh[i] = 0u;
    __syncthreads();
    const float4* H4 = (const float4*)H;
    long long stride = (long long)gridDim.x * blockDim.x;
    long long g = (long long)blockIdx.x * blockDim.x + threadIdx.x;
    for (long long i = g; i < nv4; i += stride) {
        float4 v = H4[i];
        atomicAdd(&sh[monoKey(v.x) >> 22], 1u);
        atomicAdd(&sh[monoKey(v.y) >> 22], 1u);
        atomicAdd(&sh[monoKey(v.z) >> 22], 1u);
        atomicAdd(&sh[monoKey(v.w) >> 22], 1u);
    }
    for (long long i = nv4 * 4 + g; i < n; i += stride)
        atomicAdd(&sh[monoKey(H[i]) >> 22], 1u);
    __syncthreads();
    for (int i = threadIdx.x; i < 1024; i += blockDim.x) {
        unsigned c = sh[i];
        if (c) atomicAdd(&hist1[i], c);
    }
}

// ---- generic 1024-bin boundary selector (levels 0 and 1) ------------------
// level 0: need = k          -> writes meta[0]=B1, meta[1]=cntBelow1
// level 1: need = k-meta[1]  -> writes meta[2]=B2, meta[3]=cntBelow2
__global__ __launch_bounds__(1024) void lt_select(const unsigned* __restrict__ hist,
                                                  unsigned kTotal,
                                                  unsigned* __restrict__ meta,
                                                  int level) {
    __shared__ unsigned sCnt[1024];
    __shared__ unsigned sPre[1024];
    int t = threadIdx.x;
    unsigned need = (level == 0) ? kTotal : (kTotal - meta[1]);
    unsigned c = hist[t];
    sCnt[t] = c;
    __syncthreads();
    if (t == 0) {
        unsigned run = 0;
        for (int i = 0; i < 1024; ++i) { sPre[i] = run; run += sCnt[i]; }
    }
    __syncthreads();
    unsigned before = sPre[t];
    if (before < need && before + c >= need) {
        meta[level * 2 + 0] = (unsigned)t;
        meta[level * 2 + 1] = before;
    }
}

// ---- level-2: sum elements below bin B1; histogram mid-10 bits in bin B1 --
__global__ __launch_bounds__(256) void lt_refine1(const float* __restrict__ H,
                                                  long long nv4, long long n,
                                                  const unsigned* __restrict__ meta,
                                                  unsigned* __restrict__ hist2,
                                                  double* __restrict__ sumBelow) {
    unsigned B1 = meta[0];
    double local = 0.0;
    const float4* H4 = (const float4*)H;
    long long stride = (long long)gridDim.x * blockDim.x;
    long long g = (long long)blockIdx.x * blockDim.x + threadIdx.x;
    for (long long i = g; i < nv4; i += stride) {
        float4 v = H4[i];
        float e[4] = {v.x, v.y, v.z, v.w};
#pragma unroll
        for (int j = 0; j < 4; ++j) {
            uint32_t key = monoKey(e[j]);
            uint32_t hi = key >> 22;
            if (hi < B1)       local += (double)e[j];
            else if (hi == B1) atomicAdd(&hist2[(key >> 12) & 0x3FFu], 1u);
        }
    }
    for (long long i = nv4 * 4 + g; i < n; i += stride) {
        float x = H[i];
        uint32_t key = monoKey(x);
        uint32_t hi = key >> 22;
        if (hi < B1)       local += (double)x;
        else if (hi == B1) atomicAdd(&hist2[(key >> 12) & 0x3FFu], 1u);
    }
    __shared__ double sred[256];
    sred[threadIdx.x] = local;
    __syncthreads();
    for (int off = 128; off > 0; off >>= 1) {
        if ((int)threadIdx.x < off) sred[threadIdx.x] += sred[threadIdx.x + off];
        __syncthreads();
    }
    if (threadIdx.x == 0 && sred[0] != 0.0)
        __hip_atomic_fetch_add(sumBelow, sred[0], __ATOMIC_RELAXED, __HIP_MEMORY_SCOPE_AGENT);
}

// ---- level-3: within (B1,B2), histogram+sum low-12 bits; sum below B2 -----
__global__ __launch_bounds__(256) void lt_refine2(const float* __restrict__ H,
                                                  long long nv4, long long n,
                                                  const unsigned* __restrict__ meta,
                                                  unsigned* __restrict__ hist3,
                                                  float* __restrict__ sums3,
                                                  double* __restrict__ sumBelow) {
    unsigned B1 = meta[0], B2 = meta[2];
    double local = 0.0;
    const float4* H4 = (const float4*)H;
    long long stride = (long long)gridDim.x * blockDim.x;
    long long g = (long long)blockIdx.x * blockDim.x + threadIdx.x;
    for (long long i = g; i < nv4; i += stride) {
        float4 v = H4[i];
        float e[4] = {v.x, v.y, v.z, v.w};
#pragma unroll
        for (int j = 0; j < 4; ++j) {
            uint32_t key = monoKey(e[j]);
            if ((key >> 22) == B1) {
                uint32_t mid = (key >> 12) & 0x3FFu;
                if (mid < B2) local += (double)e[j];
                else if (mid == B2) {
                    uint32_t lo = key & 0xFFFu;
                    atomicAdd(&hist3[lo], 1u);
                    __hip_atomic_fetch_add(&sums3[lo], e[j], __ATOMIC_RELAXED, __HIP_MEMORY_SCOPE_AGENT);
                }
            }
        }
    }
    for (long long i = nv4 * 4 + g; i < n; i += stride) {
        float x = H[i];
        uint32_t key = monoKey(x);
        if ((key >> 22) == B1) {
            uint32_t mid = (key >> 12) & 0x3FFu;
            if (mid < B2) local += (double)x;
            else if (mid == B2) {
                uint32_t lo = key & 0xFFFu;
                atomicAdd(&hist3[lo], 1u);
                __hip_atomic_fetch_add(&sums3[lo], x, __ATOMIC_RELAXED, __HIP_MEMORY_SCOPE_AGENT);
            }
        }
    }
    __shared__ double sred[256];
    sred[threadIdx.x] = local;
    __syncthreads();
    for (int off = 128; off > 0; off >>= 1) {
        if ((int)threadIdx.x < off) sred[threadIdx.x] += sred[threadIdx.x + off];
        __syncthreads();
    }
    if (threadIdx.x == 0 && sred[0] != 0.0)
        __hip_atomic_fetch_add(sumBelow, sred[0], __ATOMIC_RELAXED, __HIP_MEMORY_SCOPE_AGENT);
}

// ---- finalize: exact Ac from 4096-bin low histogram -----------------------
__global__ __launch_bounds__(1024) void lt_final(const unsigned* __restrict__ hist3,
                                                 const float* __restrict__ sums3,
                                                 const double* __restrict__ sumBelow,
                                                 unsigned kTotal,
                                                 unsigned* __restrict__ meta) {
    __shared__ unsigned sCnt[1024];
    __shared__ unsigned sPre[1024];
    __shared__ double   sSum[1024];
    __shared__ unsigned sBin, sTake;
    int t = threadIdx.x;
    unsigned need = kTotal - meta[1] - meta[3];
    unsigned c = 0;
#pragma unroll
    for (int j = 0; j < 4; ++j) c += hist3[t * 4 + j];
    sCnt[t] = c;
    __syncthreads();
    if (t == 0) {
        unsigned run = 0;
        for (int i = 0; i < 1024; ++i) { sPre[i] = run; run += sCnt[i]; }
    }
    __syncthreads();
    unsigned before = sPre[t];
    if (before < need && before + c >= need) {
        unsigned cum = before;
        for (int j = 0; j < 4; ++j) {
            unsigned cc = hist3[t * 4 + j];
            if (cum + cc >= need) { sBin = (unsigned)(t * 4 + j); sTake = need - cum; break; }
            cum += cc;
        }
    }
    __syncthreads();
    unsigned jb = sBin, take = sTake;
    double part = 0.0;
#pragma unroll
    for (int j = 0; j < 4; ++j) {
        unsigned b = (unsigned)(t * 4 + j);
        if (b < jb) part += (double)sums3[b];
    }
    sSum[t] = part;
    __syncthreads();
    if (t == 0) {
        double acc = *sumBelow;
        for (int i = 0; i < 1024; ++i) acc += sSum[i];
        uint32_t key = (meta[0] << 22) | (meta[2] << 12) | jb;
        acc += (double)take * (double)keyToFloat(key);
        float Ac = (float)(acc / (double)kTotal);
        ((float*)meta)[4] = Ac;
    }
}

// ---- elementwise pass: async-to-LDS double-buffered streaming -------------
#if USE_ASYNC_LDS
typedef int lt_v4i __attribute__((__vector_size__(16)));
typedef __attribute__((address_space(1))) lt_v4i* lt_gp;
typedef __attribute__((address_space(3))) lt_v4i* lt_lp;
#endif

__global__ __launch_bounds__(256) void lt_jc(const float* __restrict__ Ic,
                                             const float* __restrict__ H,
                                             float* __restrict__ out,
                                             long long nv4, long long n,
                                             const unsigned* __restrict__ meta) {
    const float Ac = ((const float*)meta)[4];
    const float4* H4 = (const float4*)H;
    const float4* I4 = (const float4*)Ic;
    float4* O4 = (float4*)out;
    long long stride = (long long)gridDim.x * blockDim.x;
    long long g = (long long)blockIdx.x * blockDim.x + threadIdx.x;

#if USE_ASYNC_LDS
    // Per-wave double-buffered LDS staging; each wave touches only its slice,
    // so no workgroup barriers are needed — only s_wait_asynccnt ordering.
    __shared__ float4 ldsH[2][8][32];
    __shared__ float4 ldsI[2][8][32];
    const int lane = threadIdx.x & 31;
    const int wave = threadIdx.x >> 5;
    long long nTiles = nv4 >> 5;                    // 32 float4 per wave-tile
    long long wgid = (long long)blockIdx.x * 8 + wave;
    long long wstride = (long long)gridDim.x * 8;

    auto issueTile = [&](long long tile, int buf) {
        long long idx = (tile << 5) + lane;
        __builtin_amdgcn_global_load_async_to_lds_b128(
            (lt_gp)(void*)(H4 + idx),
            (lt_lp)(void*)&ldsH[buf][wave][lane], 0, 0);
        __builtin_amdgcn_global_load_async_to_lds_b128(
            (lt_gp)(void*)(I4 + idx),
            (lt_lp)(void*)&ldsI[buf][wave][lane], 0, 0);
    };

    if (wgid < nTiles) {
        issueTile(wgid, 0);
        int buf = 0;
        for (long long tile = wgid; tile < nTiles; tile += wstride) {
            long long nxt = tile + wstride;
            if (nxt >= nTiles) nxt = tile;          // keep exactly 2 in flight
            issueTile(nxt, buf ^ 1);
            __builtin_amdgcn_s_wait_asynccnt(2);    // oldest 2 (current buf) done
            float4 h  = ldsH[buf][wave][lane];
            float4 ic = ldsI[buf][wave][lane];
            float4 r;
            r.x = (ic.x - Ac) / (ALPHA_C * h.x + EPS_C) + Ac;
            r.y = (ic.y - Ac) / (ALPHA_C * h.y + EPS_C) + Ac;
            r.z = (ic.z - Ac) / (ALPHA_C * h.z + EPS_C) + Ac;
            r.w = (ic.w - Ac) / (ALPHA_C * h.w + EPS_C) + Ac;
            O4[(tile << 5) + lane] = r;
            buf ^= 1;
        }
    }
    // scalar tail beyond full wave-tiles
    for (long long i = nTiles * 128 + g; i < n; i += stride) {
        float h = H[i];
        out[i] = (Ic[i] - Ac) / (ALPHA_C * h + EPS_C) + Ac;
    }
#else
    for (long long i = g; i < nv4; i += stride) {
        float4 h = H4[i];
        float4 ic = I4[i];
        float4 r;
        r.x = (ic.x - Ac) / (ALPHA_C * h.x + EPS_C) + Ac;
        r.y = (ic.y - Ac) / (ALPHA_C * h.y + EPS_C) + Ac;
        r.z = (ic.z - Ac) / (ALPHA_C * h.z + EPS_C) + Ac;
        r.w = (ic.w - Ac) / (ALPHA_C * h.w + EPS_C) + Ac;
        O4[i] = r;
    }
    for (long long i = nv4 * 4 + g; i < n; i += stride) {
        float h = H[i];
        out[i] = (Ic[i] - Ac) / (ALPHA_C * h + EPS_C) + Ac;
    }
#endif
}

// ---------------------------------------------------------------------------
extern "C" void kernel_launch(void* const* d_in, const int* in_sizes, int n_in,
                              void* d_out, int out_size, void* d_ws, size_t ws_size,
                              hipStream_t stream) {
    const float* Ic = (const float*)d_in[0];
    const float* H  = (const float*)d_in[1];
    float* out = (float*)d_out;
    long long n = (long long)in_sizes[1];
    long long nv4 = n / 4;
    unsigned k = (unsigned)(n / 1000);
    if (k < 1) k = 1;

    // workspace layout (bytes)
    uint8_t* ws = (uint8_t*)d_ws;
    unsigned* hist1   = (unsigned*)(ws + 0);        // 1024 u32
    unsigned* hist2   = (unsigned*)(ws + 4096);     // 1024 u32
    unsigned* hist3   = (unsigned*)(ws + 8192);     // 4096 u32
    float*    sums3   = (float*)   (ws + 24576);    // 4096 f32
    double*   sumBelow= (double*)  (ws + 40960);    // 1 f64
    unsigned* meta    = (unsigned*)(ws + 40968);    // B1,c1,B2,c2,Ac

    const int zWords = 10250;                       // covers everything above
    lt_zero<<<(zWords + 255) / 256, 256, 0, stream>>>((unsigned*)ws, zWords);

    int blocks = (int)((nv4 + 255) / 256);
    if (blocks < 1) blocks = 1;
    if (blocks > 8192) blocks = 8192;

    lt_hist1  <<<blocks, 256, 0, stream>>>(H, nv4, n, hist1);
    lt_select <<<1, 1024, 0, stream>>>(hist1, k, meta, 0);
    lt_refine1<<<blocks, 256, 0, stream>>>(H, nv4, n, meta, hist2, sumBelow);
    lt_select <<<1, 1024, 0, stream>>>(hist2, k, meta, 1);
    lt_refine2<<<blocks, 256, 0, stream>>>(H, nv4, n, meta, hist3, sums3, sumBelow);
    lt_final  <<<1, 1024, 0, stream>>>(hist3, sums3, sumBelow, k, meta);

    int jblocks = 4096;
    if ((long long)jblocks * 256 > n && n > 0) {
        jblocks = (int)((n + 255) / 256);
        if (jblocks < 1) jblocks = 1;
    }
    lt_jc<<<jblocks, 256, 0, stream>>>(Ic, H, out, nv4, n, meta);
}